// KDCDataGen_50912542327117
// MI455X (gfx1250) — compile-verified
//
#include <hip/hip_runtime.h>
#include <math.h>

// ---------------------------------------------------------------------------
// KDC pyrazine Lindblad propagator on gfx1250 (MI455X).
//   D = 48 (3 el x 16 vib), Liouville dim N = D^2 = 2304.
//   X  = (DT/(HBAR*2^S)) * (Lc + g*Ld0)        (built elementwise on device)
//   e^X via Horner Taylor (12 terms), U = (e^X)^(2^S), S = 8
//   U32 = U^32; propagate in panels of 32 steps: B_{j+1} = U32 * B_j
//   pops_k(t) = sum_v Re rho[(k*16+v)*49]  (diagonal trace)
// Complex GEMMs: split-plane FP32 WMMA (v_wmma_f32_16x16x4_f32) with the
// Gauss 3-multiply trick (P1=aR*bR, P2=aI*bI, Ks=(aR+aI)(bR+bI)):
//   CR = P1 - P2, CI = Ks - P1 - P2   -> 25% fewer matrix ops.
// ---------------------------------------------------------------------------

typedef __attribute__((ext_vector_type(2))) float v2f;
typedef __attribute__((ext_vector_type(4))) float v4f;
typedef __attribute__((ext_vector_type(8))) float v8f;

#define NN    2304          // Liouville dimension
#define DD    48
#define S_SQ  8             // scaling-and-squaring exponent
#define K_TAY 12            // Taylor order

__device__ __forceinline__ v8f wmma_f32x4(v2f a, v2f b, v8f c) {
  // 8 args: (neg_a, A, neg_b, B, c_mod, C, reuse_a, reuse_b); f32 has no A/B neg.
  return __builtin_amdgcn_wmma_f32_16x16x4_f32(false, a, false, b, (short)0, c,
                                               false, false);
}

// ----------------------------- small builders ------------------------------

__device__ __forceinline__ float Qe(int m, int n) {
  // Q = (a + a^dag)/sqrt(2), truncated 4x4
  if (m + 1 == n) return sqrtf((float)n) * 0.70710678118f;
  if (n + 1 == m) return sqrtf((float)m) * 0.70710678118f;
  return 0.f;
}

__global__ void build_H(float* __restrict__ H) {
  int idx = blockIdx.x * blockDim.x + threadIdx.x;
  if (idx >= DD * DD) return;
  int a = idx / DD, c = idx % DD;
  int e = a / 16, r6 = (a / 4) % 4, r10 = a % 4;
  int f = c / 16, s6 = (c / 4) % 4, s10 = c % 4;
  const float CM2EV = 0.00012398419f;
  float h = 0.f;
  bool d6 = (r6 == s6), d10 = (r10 == s10);
  if (e == f && d6 && d10) {
    if (e == 1) h += 3.995f;
    if (e == 2) h += 4.9183f;
    h += 596.0f * CM2EV * (float)r6 + 919.0f * CM2EV * (float)r10;
  }
  if (e == f && d10) {
    if (e == 1) h += -0.0964f * Qe(r6, s6);
    if (e == 2) h += 0.1193f * Qe(r6, s6);
  }
  if (d6 && ((e == 1 && f == 2) || (e == 2 && f == 1))) {
    float q2 = 0.f;
    for (int p = 0; p < 4; ++p) q2 += Qe(r10, p) * Qe(p, s10);
    h += 0.1825f * Qe(r10, s10) + (-0.018f) * q2;
  }
  H[idx] = h;
}

// X[(a,b),(c,d)] = scale * [ -i(d_ac H[b,d] - d_bd H[a,c])
//                            + g (Lj[a,c]Lj[b,d] - .5 d_ac P2[b,d] - .5 P2[a,c] d_bd) ]
__global__ void build_X(const float* __restrict__ H, const float* __restrict__ log_g,
                        float* __restrict__ XR, float* __restrict__ XI, float scale) {
  int idx = blockIdx.x * blockDim.x + threadIdx.x;
  if (idx >= NN * NN) return;
  int i = idx / NN, j = idx % NN;
  int a = i / DD, b = i % DD;
  int c = j / DD, d = j % DD;
  float g = expf(log_g[0]);
  float im = 0.f;
  if (a == c) im -= H[b * DD + d];
  if (b == d) im += H[a * DD + c];
  int ea = a / 16, eb = b / 16, ec = c / 16, ed = d / 16;
  float lac = (ea == 1 && ec == 2 && (a % 16) == (c % 16)) ? 1.f : 0.f;
  float lbd = (eb == 1 && ed == 2 && (b % 16) == (d % 16)) ? 1.f : 0.f;
  float re = lac * lbd;
  if (a == c && b == d) {
    if (eb == 2) re -= 0.5f;
    if (ea == 2) re -= 0.5f;
  }
  re *= g;
  XR[idx] = re * scale;
  XI[idx] = im * scale;
}

__global__ void init_identity(float* __restrict__ AR, float* __restrict__ AI) {
  int idx = blockIdx.x * blockDim.x + threadIdx.x;
  if (idx >= NN * NN) return;
  int i = idx / NN, j = idx % NN;
  AR[idx] = (i == j) ? 1.f : 0.f;
  AI[idx] = 0.f;
}

// panel B: 2304 x 32 row-major (ld = 32); rho0 = e_{1568} (a=b=32 -> 32*49=1568)
__global__ void init_B(float* __restrict__ BR, float* __restrict__ BI) {
  int idx = blockIdx.x * blockDim.x + threadIdx.x;
  if (idx >= NN * 32) return;
  BR[idx] = (idx == 1568 * 32) ? 1.f : 0.f;
  BI[idx] = 0.f;
}

// ------------------------- complex WMMA GEMM -------------------------------
// C[MxNn] = alpha * A[MxK] * B[KxNn]  (+ I on the real diagonal if addI)
// Split planes, row-major, ld = width. Requires K % 32 == 0 (true here: 2304).
// Block = 256 threads (8 waves), block tile 128x64, K-tile 32.
// Waves arranged 4x2; each wave owns a 32x32 tile = 2x2 WMMA subtiles with
// three f32 accumulators each (Gauss complex multiply).

#define BMT 128
#define BNT 64
#define BKT 32
#define LDA_S 36   // 32 + 4 pad: keeps 16B alignment, spreads banks
#define LDB_S 68   // 64 + 4 pad

__global__ __launch_bounds__(256)
void cgemm_wmma(const float* __restrict__ AR, const float* __restrict__ AI,
                const float* __restrict__ BR, const float* __restrict__ BI,
                float* __restrict__ CR, float* __restrict__ CI,
                int M, int Nn, int K, float alpha, int addI) {
  __shared__ float smem[2 * BMT * LDA_S + 2 * BKT * LDB_S];
  float* AsR = smem;                       // 128 x 36
  float* AsI = AsR + BMT * LDA_S;
  float* BsR = AsI + BMT * LDA_S;          // 32 x 68
  float* BsI = BsR + BKT * LDB_S;

  const int row0 = blockIdx.y * BMT;
  const int col0 = blockIdx.x * BNT;
  const int tid = threadIdx.x;
  const int wave = tid >> 5;               // 0..7
  const int lane = tid & 31;
  const int wr = wave >> 1;                // wave row 0..3  (32 rows each)
  const int wc = wave & 1;                 // wave col 0..1  (32 cols each)
  const int lh = lane >> 4;                // K-half / M-half selector
  const int lm = lane & 15;

  const v8f zero = {0.f, 0.f, 0.f, 0.f, 0.f, 0.f, 0.f, 0.f};
  v8f P1[2][2] = {{zero, zero}, {zero, zero}};   // aR*bR
  v8f P2[2][2] = {{zero, zero}, {zero, zero}};   // aI*bI
  v8f Ks[2][2] = {{zero, zero}, {zero, zero}};   // (aR+aI)*(bR+bI)

  for (int k0 = 0; k0 < K; k0 += BKT) {
    // ---- stage A: 128x32 per plane as b128 loads (4 float4 per thread) ----
#pragma unroll
    for (int it = 0; it < 4; ++it) {
      int e = tid + it * 256;              // 0..1023
      int r = e >> 3, q = (e & 7) << 2;    // row 0..127, col q..q+3
      int gr = row0 + r;
      v4f vr = {0.f, 0.f, 0.f, 0.f}, vi = {0.f, 0.f, 0.f, 0.f};
      if (gr < M) {                        // k0+q < K guaranteed (K % 32 == 0)
        size_t o = (size_t)gr * K + (k0 + q);
        vr = *(const v4f*)&AR[o];
        vi = *(const v4f*)&AI[o];
      }
      *(v4f*)&AsR[r * LDA_S + q] = vr;
      *(v4f*)&AsI[r * LDA_S + q] = vi;
    }
    // ---- stage B: 32x64 per plane (2 float4 per thread) ----
#pragma unroll
    for (int it = 0; it < 2; ++it) {
      int e = tid + it * 256;              // 0..511
      int kk = e >> 4, q = (e & 15) << 2;  // k-row 0..31, col q..q+3
      int gk = k0 + kk, gc = col0 + q;
      v4f vr = {0.f, 0.f, 0.f, 0.f}, vi = {0.f, 0.f, 0.f, 0.f};
      if (gc + 3 < Nn) {
        size_t o = (size_t)gk * Nn + gc;
        vr = *(const v4f*)&BR[o];
        vi = *(const v4f*)&BI[o];
      } else {
        for (int z = 0; z < 4; ++z)
          if (gc + z < Nn) {
            size_t o = (size_t)gk * Nn + gc + z;
            ((float*)&vr)[z] = BR[o];
            ((float*)&vi)[z] = BI[o];
          }
      }
      *(v4f*)&BsR[kk * LDB_S + q] = vr;
      *(v4f*)&BsI[kk * LDB_S + q] = vi;
    }

    // ---- prefetch next K-tile into cache (global_prefetch_b8) ----
    if (k0 + BKT < K) {
      int r = tid >> 1;                    // 0..127
      const float* pa = (tid & 1) ? AI : AR;
      if (row0 + r < M)
        __builtin_prefetch(&pa[(size_t)(row0 + r) * K + (k0 + BKT)], 0, 1);
      if (tid < 64) {
        const float* pb = (tid & 1) ? BI : BR;
        int kp = tid >> 1;                 // 0..31
        if (col0 < Nn)
          __builtin_prefetch(&pb[(size_t)(k0 + BKT + kp) * Nn + col0], 0, 1);
      }
    }
    __syncthreads();

    // ---- compute: 8 K-steps x 12 WMMA per wave ----
#pragma unroll
    for (int kk = 0; kk < BKT; kk += 4) {
      int ak = kk + (lh << 1);             // lanes 0-15: K={kk,kk+1}; 16-31: +2
      v2f aR[2], aI[2], aS[2];
#pragma unroll
      for (int ri = 0; ri < 2; ++ri) {
        int arow = wr * 32 + ri * 16 + lm;
        aR[ri].x = AsR[arow * LDA_S + ak];
        aR[ri].y = AsR[arow * LDA_S + ak + 1];
        aI[ri].x = AsI[arow * LDA_S + ak];
        aI[ri].y = AsI[arow * LDA_S + ak + 1];
        aS[ri] = aR[ri] + aI[ri];
      }
      v2f bR[2], bI[2], bS[2];
#pragma unroll
      for (int ci = 0; ci < 2; ++ci) {
        int bn = wc * 32 + ci * 16 + lm;
        bR[ci].x = BsR[ak * LDB_S + bn];
        bR[ci].y = BsR[(ak + 1) * LDB_S + bn];
        bI[ci].x = BsI[ak * LDB_S + bn];
        bI[ci].y = BsI[(ak + 1) * LDB_S + bn];
        bS[ci] = bR[ci] + bI[ci];
      }
#pragma unroll
      for (int ri = 0; ri < 2; ++ri)
#pragma unroll
        for (int ci = 0; ci < 2; ++ci) {
          P1[ri][ci] = wmma_f32x4(aR[ri], bR[ci], P1[ri][ci]);
          P2[ri][ci] = wmma_f32x4(aI[ri], bI[ci], P2[ri][ci]);
          Ks[ri][ci] = wmma_f32x4(aS[ri], bS[ci], Ks[ri][ci]);
        }
    }
    __syncthreads();
  }

  // epilogue: C layout — VGPR r: lanes 0-15 => M=r, lanes 16-31 => M=r+8
#pragma unroll
  for (int ri = 0; ri < 2; ++ri)
#pragma unroll
    for (int ci = 0; ci < 2; ++ci) {
      int col = col0 + wc * 32 + ci * 16 + lm;
#pragma unroll
      for (int r = 0; r < 8; ++r) {
        int row = row0 + wr * 32 + ri * 16 + (lh << 3) + r;
        if (row < M && col < Nn) {
          float p1 = P1[ri][ci][r], p2 = P2[ri][ci][r];
          float cr = alpha * (p1 - p2);
          float cim = alpha * (Ks[ri][ci][r] - p1 - p2);
          if (addI && row == col) cr += 1.f;
          size_t o = (size_t)row * Nn + col;
          CR[o] = cr;
          CI[o] = cim;
        }
      }
    }
}

// --------------------- serial matvec (panel seeding) -----------------------
// y = U x (complex); x = column csrc of B, y = column cdst. 8 rows per block.
__global__ __launch_bounds__(256)
void cmatvec_col(const float* __restrict__ UR, const float* __restrict__ UI,
                 float* __restrict__ BR, float* __restrict__ BI,
                 int csrc, int cdst) {
  int wave = threadIdx.x >> 5;
  int lane = threadIdx.x & 31;
  int row = blockIdx.x * 8 + wave;
  if (row >= NN) return;
  float ar = 0.f, ai = 0.f;
  for (int k = lane; k < NN; k += 32) {
    float ur = UR[(size_t)row * NN + k], ui = UI[(size_t)row * NN + k];
    float xr = BR[k * 32 + csrc], xi = BI[k * 32 + csrc];
    ar += ur * xr - ui * xi;
    ai += ur * xi + ui * xr;
  }
#pragma unroll
  for (int off = 16; off > 0; off >>= 1) {
    ar += __shfl_down(ar, off, 32);
    ai += __shfl_down(ai, off, 32);
  }
  if (lane == 0) {
    BR[row * 32 + cdst] = ar;
    BI[row * 32 + cdst] = ai;
  }
}

// --------------------------- trace populations -----------------------------
__global__ void pops_kernel(const float* __restrict__ BR, float* __restrict__ out,
                            int chunk, int n_steps) {
  int tid = threadIdx.x;
  if (tid >= 96) return;
  int k = tid >> 5, c = tid & 31;
  int t = chunk * 32 + c;
  if (t >= n_steps) return;
  float s = 0.f;
#pragma unroll
  for (int v = 0; v < 16; ++v) {
    int a = k * 16 + v;          // diagonal rho[a,a] -> flat a*49
    s += BR[(a * 49) * 32 + c];
  }
  out[k * n_steps + t] = s;
}

// ------------------------------ orchestration ------------------------------

static inline void launch_cgemm(const float* AR, const float* AI,
                                const float* BR, const float* BI,
                                float* CR, float* CI, int M, int Nn, int K,
                                float alpha, int addI, hipStream_t s) {
  dim3 grid((Nn + BNT - 1) / BNT, (M + BMT - 1) / BMT);
  cgemm_wmma<<<grid, 256, 0, s>>>(AR, AI, BR, BI, CR, CI, M, Nn, K, alpha, addI);
}

extern "C" void kernel_launch(void* const* d_in, const int* in_sizes, int n_in,
                              void* d_out, int out_size, void* d_ws, size_t ws_size,
                              hipStream_t stream) {
  const float* log_g = (const float*)d_in[0];
  float* out = (float*)d_out;
  const int n_steps = out_size / 3;

  // workspace layout (floats); requires ~129 MB of scratch
  const size_t N2 = (size_t)NN * NN;
  float* ws = (float*)d_ws;
  float* H = ws;                         // 2304 floats
  float* base = ws + 4096;
  float* P[3][2];
  for (int i = 0; i < 3; ++i)
    for (int p = 0; p < 2; ++p) P[i][p] = base + (size_t)(2 * i + p) * N2;
  float* Bbase = base + 6 * N2;
  const size_t BP = (size_t)NN * 32;
  float* B0R = Bbase;
  float* B0I = Bbase + BP;
  float* B1R = Bbase + 2 * BP;
  float* B1I = Bbase + 3 * BP;

  const float scale = (1.0f / 0.6582119569f) / (float)(1 << S_SQ); // DT/(HBAR*2^S)
  const int g1 = (DD * DD + 255) / 256;
  const int g2 = (int)((N2 + 255) / 256);

  build_H<<<g1, 256, 0, stream>>>(H);
  build_X<<<g2, 256, 0, stream>>>(H, log_g, P[0][0], P[0][1], scale);
  init_identity<<<g2, 256, 0, stream>>>(P[1][0], P[1][1]);

  // Horner Taylor: Acc <- I + (X/k) * Acc, k = K_TAY..1
  int acc = 1, tmp = 2;
  for (int k = K_TAY; k >= 1; --k) {
    launch_cgemm(P[0][0], P[0][1], P[acc][0], P[acc][1], P[tmp][0], P[tmp][1],
                 NN, NN, NN, 1.0f / (float)k, 1, stream);
    int t = acc; acc = tmp; tmp = t;
  }

  // squarings: U = (e^X)^(2^S)
  int cur = acc;
  for (int i = 0; i < S_SQ; ++i) {
    int dst = (cur + 1) % 3;
    launch_cgemm(P[cur][0], P[cur][1], P[cur][0], P[cur][1], P[dst][0], P[dst][1],
                 NN, NN, NN, 1.f, 0, stream);
    cur = dst;
  }
  const int u = cur;

  // seed panel B0 = [rho0, U rho0, ..., U^31 rho0]
  init_B<<<(int)((BP + 255) / 256), 256, 0, stream>>>(B0R, B0I);
  for (int c = 0; c < 31; ++c)
    cmatvec_col<<<NN / 8, 256, 0, stream>>>(P[u][0], P[u][1], B0R, B0I, c, c + 1);

  // U^32 (stream-ordered after the matvecs, so overwriting U's buffer is safe)
  for (int i = 0; i < 5; ++i) {
    int dst = (cur + 1) % 3;
    launch_cgemm(P[cur][0], P[cur][1], P[cur][0], P[cur][1], P[dst][0], P[dst][1],
                 NN, NN, NN, 1.f, 0, stream);
    cur = dst;
  }
  const int u32 = cur;

  // panel propagation: pops(B_j); B_{j+1} = U^32 * B_j (L2-resident GEMM)
  int nchunks = (n_steps + 31) / 32;
  float *CRp = B0R, *CIp = B0I, *NRp = B1R, *NIp = B1I;
  for (int j = 0; j < nchunks; ++j) {
    pops_kernel<<<1, 128, 0, stream>>>(CRp, out, j, n_steps);
    if (j + 1 < nchunks) {
      launch_cgemm(P[u32][0], P[u32][1], CRp, CIp, NRp, NIp, NN, 32, NN, 1.f, 0,
                   stream);
      float* t;
      t = CRp; CRp = NRp; NRp = t;
      t = CIp; CIp = NIp; NIp = t;
    }
  }
}